// LargeMarginSoftmaxLoss_75359496176066
// MI455X (gfx1250) — compile-verified
//
#include <hip/hip_runtime.h>
#include <math.h>

typedef float v2f __attribute__((ext_vector_type(2)));
typedef float v8f __attribute__((ext_vector_type(8)));

#define N_ROWS 4096
#define E_DIM  512
#define C_COLS 50257
#define EPSN   1e-12f
// cos(4.0), sin(4.0)
#define COS_M  (-0.65364362086361194f)
#define SIN_M  (-0.75680249530792820f)

#define BM   64          // block M tile
#define BC   256         // block C tile
#define KC   32          // K chunk staged in LDS
#define LSTR 36          // padded LDS row stride (dwords): conflict-free, 16B aligned
#define A_FLOATS (BM * LSTR)    // 2304
#define B_FLOATS (BC * LSTR)    // 9216
#define SMEM_FLOATS (2 * A_FLOATS + 2 * B_FLOATS)   // 23040 floats = 92160 B

// ---------------------------------------------------------------------------
// Kernel 1: inverse L2 norms of x rows and weight rows; also zero sumexp.
// One wave (32 lanes) per row of 512 floats.
// ---------------------------------------------------------------------------
__global__ void __launch_bounds__(256)
lm_norms_kernel(const float* __restrict__ x, const float* __restrict__ w,
                float* __restrict__ rnx, float* __restrict__ rnw,
                float* __restrict__ sumexp)
{
    const int gw   = (int)((blockIdx.x * blockDim.x + threadIdx.x) >> 5);
    const int lane = (int)(threadIdx.x & 31);
    const int total = N_ROWS + C_COLS;
    if (gw >= total) return;

    const float* row = (gw < N_ROWS) ? (x + (size_t)gw * E_DIM)
                                     : (w + (size_t)(gw - N_ROWS) * E_DIM);
    float s = 0.f;
#pragma unroll
    for (int j = 0; j < E_DIM / 128; ++j) {
        float4 v = *(const float4*)(row + j * 128 + lane * 4);
        s += v.x * v.x + v.y * v.y + v.z * v.z + v.w * v.w;
    }
#pragma unroll
    for (int m = 16; m >= 1; m >>= 1) s += __shfl_xor(s, m, 32);

    if (lane == 0) {
        float rn = 1.0f / fmaxf(sqrtf(s), EPSN);
        if (gw < N_ROWS) { rnx[gw] = rn; sumexp[gw] = 0.f; }
        else             { rnw[gw - N_ROWS] = rn; }
    }
}

// ---------------------------------------------------------------------------
// Async global -> LDS copy of one 16-byte chunk (CDNA5, tracked by ASYNCcnt).
// ---------------------------------------------------------------------------
__device__ __forceinline__ void async_load16(unsigned lds_byte_off, const float* gptr)
{
    asm volatile("global_load_async_to_lds_b128 %0, %1, off"
                 :: "v"(lds_byte_off), "v"((unsigned long long)(uintptr_t)gptr)
                 : "memory");
}

__device__ __forceinline__ void wait_asynccnt0()
{
    asm volatile("s_wait_asynccnt 0x0" ::: "memory");
}

// Stage one (BM x KC) A tile + (BC x KC) B tile into LDS asynchronously.
// 16B chunks: A = 512 chunks, B = 2048 chunks; 256 threads.
__device__ __forceinline__ void stage_tile_async(
    const float* __restrict__ x, const float* __restrict__ w,
    int blockM, int blockC, int kb,
    unsigned aBase, unsigned bBase, int wid, int lane)
{
#pragma unroll
    for (int j = 0; j < 2; ++j) {                       // A: 2 chunks/lane
        const int q = wid * 64 + j * 32 + lane;         // 0..511
        const int r = q >> 3, kq = q & 7;
        async_load16(aBase + (unsigned)(r * (LSTR * 4) + kq * 16),
                     x + (size_t)(blockM + r) * E_DIM + kb + kq * 4);
    }
#pragma unroll
    for (int j = 0; j < 8; ++j) {                       // B: 8 chunks/lane
        const int q = wid * 256 + j * 32 + lane;        // 0..2047
        const int c = q >> 3, kq = q & 7;
        const int gc = min(blockC + c, C_COLS - 1);
        async_load16(bBase + (unsigned)(c * (LSTR * 4) + kq * 16),
                     w + (size_t)gc * E_DIM + kb + kq * 4);
    }
}

// ---------------------------------------------------------------------------
// Kernel 2: fused cos = (x/||x||) @ (w/||w||)^T with CosFace margin epilogue.
// Block = 256 threads = 8 waves; tile 64(M) x 256(C); double-buffered async
// K staging (KC=32). Each wave: 32x64 output = 8 x v8f accumulators via
// V_WMMA_F32_16X16X4_F32 (2 A-frags x 4 B-frags per k-step).
// Logits are cosines (<= 1) so logsumexp needs no max pass: one streaming
// atomic-accumulated sum of exp per row.
// ---------------------------------------------------------------------------
__global__ void __launch_bounds__(256)
lm_gemm_kernel(const float* __restrict__ x, const float* __restrict__ w,
               const int* __restrict__ label,
               const float* __restrict__ rnx, const float* __restrict__ rnw,
               float* __restrict__ sumexp, float* __restrict__ target)
{
    __shared__ float smem[SMEM_FLOATS];

    const int tid  = (int)threadIdx.x;
    const int lane = tid & 31;
    const int wid  = tid >> 5;
    const int l16  = lane & 15;
    const int hi   = lane >> 4;      // half-wave: 0 or 1
    const int wRow = wid & 1;        // 2 wave rows  -> 64 M (32 per wave)
    const int wCol = wid >> 1;       // 4 wave cols  -> 256 C (64 per wave)

    const int blockM = (int)blockIdx.y * BM;
    const int blockC = (int)blockIdx.x * BC;

    // LDS layout: [A0 | A1 | B0 | B1]; 32-bit LDS byte offsets for async ops.
    // (Computed arithmetically per use -- no const arrays of LDS-derived
    //  pointers, which lower to unsupported static initializers.)
    const unsigned ldsBase = (unsigned)(uintptr_t)&smem[0];

    v8f acc[2][4];
#pragma unroll
    for (int t = 0; t < 2; ++t)
#pragma unroll
        for (int s = 0; s < 4; ++s)
#pragma unroll
            for (int i = 0; i < 8; ++i) acc[t][s][i] = 0.f;

    // prologue: stage tile 0 into buffer 0
    stage_tile_async(x, w, blockM, blockC, 0,
                     ldsBase,
                     ldsBase + 2u * A_FLOATS * 4u,
                     wid, lane);

    int buf = 0;
    const int NIT = E_DIM / KC;     // 16
    for (int it = 0; it < NIT; ++it) {
        wait_asynccnt0();           // my async loads for this tile landed
        __syncthreads();            // everyone's loads landed; prev compute done

        if (it + 1 < NIT) {         // stream next tile into the other buffer
            const int nb = buf ^ 1;
            stage_tile_async(x, w, blockM, blockC, (it + 1) * KC,
                             ldsBase + (unsigned)nb * (A_FLOATS * 4u),
                             ldsBase + 2u * A_FLOATS * 4u + (unsigned)nb * (B_FLOATS * 4u),
                             wid, lane);
        }

        const float* As = smem + buf * A_FLOATS;
        const float* Bs = smem + 2 * A_FLOATS + buf * B_FLOATS;
#pragma unroll
        for (int k = 0; k < KC; k += 4) {
            v2f a[2];
#pragma unroll
            for (int t = 0; t < 2; ++t) {
                const int aoff = (wRow * 32 + t * 16 + l16) * LSTR + k + 2 * hi;
                a[t][0] = As[aoff]; a[t][1] = As[aoff + 1];
            }
            v2f b[4];
#pragma unroll
            for (int s = 0; s < 4; ++s) {
                const int boff = (wCol * 64 + s * 16 + l16) * LSTR + k + 2 * hi;
                b[s][0] = Bs[boff]; b[s][1] = Bs[boff + 1];
            }
#pragma unroll
            for (int t = 0; t < 2; ++t)
#pragma unroll
                for (int s = 0; s < 4; ++s)
                    acc[t][s] = __builtin_amdgcn_wmma_f32_16x16x4_f32(
                        false, a[t], false, b[s], (short)0, acc[t][s], false, false);
        }
        __syncthreads();            // done reading buf before it is refilled
        buf ^= 1;
    }

    // ---- epilogue ----
    float rw[4];
#pragma unroll
    for (int s = 0; s < 4; ++s) {
        const int gc = blockC + wCol * 64 + s * 16 + l16;
        rw[s] = rnw[min(gc, C_COLS - 1)];
    }

#pragma unroll
    for (int t = 0; t < 2; ++t) {
        float rx[8]; int lab[8];
#pragma unroll
        for (int i = 0; i < 8; ++i) {
            const int gm = blockM + wRow * 32 + t * 16 + hi * 8 + i;  // C/D: VGPR i -> M = i+8*hi
            rx[i]  = rnx[gm];
            lab[i] = label[gm];
        }
        float rowAcc[8];
#pragma unroll
        for (int i = 0; i < 8; ++i) rowAcc[i] = 0.f;

#pragma unroll
        for (int s = 0; s < 4; ++s) {
            const int gc = blockC + wCol * 64 + s * 16 + l16;
            const bool valid = (gc < C_COLS);
#pragma unroll
            for (int i = 0; i < 8; ++i) {
                float c = acc[t][s][i] * rx[i] * rw[s];
                c = fminf(fmaxf(c, -1.f), 1.f);
                float logit = c;
                if (valid && gc == lab[i]) {
                    float sint = sqrtf(fmaxf(1.f - c * c, 0.f));
                    logit = c * COS_M - sint * SIN_M;
                    const int gm = blockM + wRow * 32 + t * 16 + hi * 8 + i;
                    target[gm] = logit;           // unique writer per row
                }
                rowAcc[i] += valid ? __expf(logit) : 0.f;
            }
        }

#pragma unroll
        for (int i = 0; i < 8; ++i) {
            float v = rowAcc[i];
            v += __shfl_xor(v, 1, 32);
            v += __shfl_xor(v, 2, 32);
            v += __shfl_xor(v, 4, 32);
            v += __shfl_xor(v, 8, 32);
            if (l16 == 0) {
                const int gm = blockM + wRow * 32 + t * 16 + hi * 8 + i;
                atomicAdd(&sumexp[gm], v);
            }
        }
    }
}

// ---------------------------------------------------------------------------
// Kernel 3: loss = mean( log(sumexp[n]) - target[n] )
// ---------------------------------------------------------------------------
__global__ void __launch_bounds__(256)
lm_loss_kernel(const float* __restrict__ sumexp, const float* __restrict__ target,
               float* __restrict__ out)
{
    __shared__ float red[8];
    const int tid = (int)threadIdx.x;
    float s = 0.f;
    for (int n = tid; n < N_ROWS; n += 256)
        s += __logf(sumexp[n]) - target[n];
#pragma unroll
    for (int m = 16; m >= 1; m >>= 1) s += __shfl_xor(s, m, 32);
    if ((tid & 31) == 0) red[tid >> 5] = s;
    __syncthreads();
    if (tid == 0) {
        float t = 0.f;
#pragma unroll
        for (int i = 0; i < 8; ++i) t += red[i];
        out[0] = t / (float)N_ROWS;
    }
}

// ---------------------------------------------------------------------------
extern "C" void kernel_launch(void* const* d_in, const int* in_sizes, int n_in,
                              void* d_out, int out_size, void* d_ws, size_t ws_size,
                              hipStream_t stream)
{
    (void)in_sizes; (void)n_in; (void)out_size; (void)ws_size;

    const float* x     = (const float*)d_in[0];
    const int*   label = (const int*)d_in[1];
    const float* w     = (const float*)d_in[2];
    float*       out   = (float*)d_out;

    float* rnx    = (float*)d_ws;           // 4096
    float* rnw    = rnx + N_ROWS;           // 50257 (padded to 50304)
    float* sumexp = rnw + 50304;            // 4096
    float* target = sumexp + N_ROWS;        // 4096

    const int totalRows = N_ROWS + C_COLS;
    const int nb = (totalRows + 7) / 8;
    lm_norms_kernel<<<nb, 256, 0, stream>>>(x, w, rnx, rnw, sumexp);

    dim3 grid((C_COLS + BC - 1) / BC, N_ROWS / BM);
    lm_gemm_kernel<<<grid, 256, 0, stream>>>(x, w, label, rnx, rnw, sumexp, target);

    lm_loss_kernel<<<1, 256, 0, stream>>>(sumexp, target, out);
}